// MSARowAttentionWithPairBias_16655883174274
// MI455X (gfx1250) — compile-verified
//
#include <hip/hip_runtime.h>
#include <math.h>

typedef float v2f __attribute__((ext_vector_type(2)));
typedef float v8f __attribute__((ext_vector_type(8)));

#define S_  256
#define R_  384
#define CM_ 256
#define CZ_ 128
#define C_  32
#define H_  8
#define SR_ (S_*R_)   // 98304
#define P_  (R_*R_)   // 147456

__device__ __forceinline__ float wave_sum(float v) {
#pragma unroll
  for (int m = 16; m >= 1; m >>= 1) v += __shfl_xor(v, m, 32);
  return v;
}
__device__ __forceinline__ float wave_max(float v) {
#pragma unroll
  for (int m = 16; m >= 1; m >>= 1) v = fmaxf(v, __shfl_xor(v, m, 32));
  return v;
}

// LDS byte offset of a __shared__ pointer (flat LDS aperture: low 32 bits).
__device__ __forceinline__ unsigned lds_off(const void* p) {
  return (unsigned)(unsigned long long)p;
}

// ---------------------------------------------------------------------------
// K1: per-row stats of pair (head-invariant): phat = (pair - mu) * rsqrt(var+eps)
// ---------------------------------------------------------------------------
__global__ void pair_stats_kernel(const float* __restrict__ pair, float* __restrict__ phat) {
  const int wave = threadIdx.x >> 5, lane = threadIdx.x & 31;
  const int row  = blockIdx.x * 8 + wave;
  const float* x = pair + (size_t)row * CZ_;
  float v[4]; float s = 0.f;
#pragma unroll
  for (int j = 0; j < 4; ++j) { v[j] = x[lane + 32*j]; s += v[j]; }
  const float mu = wave_sum(s) * (1.0f / CZ_);
  float q = 0.f;
#pragma unroll
  for (int j = 0; j < 4; ++j) { float d = v[j] - mu; q += d * d; }
  const float rs = rsqrtf(wave_sum(q) * (1.0f / CZ_) + 1e-5f);
  float* y = phat + (size_t)row * CZ_;
#pragma unroll
  for (int j = 0; j < 4; ++j) y[lane + 32*j] = (v[j] - mu) * rs;
}

// ---------------------------------------------------------------------------
// K2: per-head small prep: wcat (q|k|v|g weights, 128x256), wb_eff (256x128),
//     bias_eff (256), wo_eff (256x32 slice of out_w)
// ---------------------------------------------------------------------------
__global__ void head_prep_kernel(int h,
    const float* __restrict__ wq, const float* __restrict__ wk,
    const float* __restrict__ wv, const float* __restrict__ wg,
    const float* __restrict__ npw, const float* __restrict__ npb,
    const float* __restrict__ wb, const float* __restrict__ out_w,
    float* __restrict__ wcat, float* __restrict__ wb_eff,
    float* __restrict__ bias_eff, float* __restrict__ wo_eff) {
  const int bid = blockIdx.x, tid = threadIdx.x;
  if (bid < 128) {                       // wcat: 32768 elems
    const int idx = bid * 256 + tid;
    const int n = idx >> 8, k = idx & 255;
    const float* src = (n < 32) ? wq : (n < 64) ? wk : (n < 96) ? wv : wg;
    wcat[idx] = src[(size_t)h * C_ * CM_ + (size_t)(n & 31) * CM_ + k];
  } else if (bid < 256) {                // wb_eff: 32768 elems
    const int idx = (bid - 128) * 256 + tid;
    const int sI = idx >> 7, c = idx & 127;
    wb_eff[idx] = wb[(size_t)h * S_ * CZ_ + (size_t)sI * CZ_ + c] * npw[h * CZ_ + c];
  } else if (bid < 288) {                // wo_eff: 8192 elems
    const int idx = (bid - 256) * 256 + tid;
    const int m = idx >> 5, c = idx & 31;
    wo_eff[idx] = out_w[(size_t)m * (C_ * H_) + h * C_ + c];
  } else {                               // bias_eff: 256 dots over 128
    float acc = 0.f;
    for (int c = 0; c < CZ_; ++c)
      acc += npb[h * CZ_ + c] * wb[(size_t)h * S_ * CZ_ + (size_t)tid * CZ_ + c];
    bias_eff[tid] = acc;
  }
}

// ---------------------------------------------------------------------------
// K3: msa layernorm (compounds across heads): dst = LN(src)*w + b
// ---------------------------------------------------------------------------
__global__ void ln_msa_kernel(const float* __restrict__ src, float* __restrict__ dst,
                              const float* __restrict__ w, const float* __restrict__ b) {
  const int wave = threadIdx.x >> 5, lane = threadIdx.x & 31;
  const int row  = blockIdx.x * 8 + wave;
  const float* x = src + (size_t)row * CM_;
  float v[8]; float s = 0.f;
#pragma unroll
  for (int j = 0; j < 8; ++j) { v[j] = x[lane + 32*j]; s += v[j]; }
  const float mu = wave_sum(s) * (1.0f / CM_);
  float q = 0.f;
#pragma unroll
  for (int j = 0; j < 8; ++j) { float d = v[j] - mu; q += d * d; }
  const float rs = rsqrtf(wave_sum(q) * (1.0f / CM_) + 1e-5f);
  float* y = dst + (size_t)row * CM_;
#pragma unroll
  for (int j = 0; j < 8; ++j) {
    const int c = lane + 32*j;
    y[c] = (v[j] - mu) * rs * w[c] + b[c];
  }
}

// ---------------------------------------------------------------------------
// Generic WMMA fp32 GEMM: Y(MxN) = X(MxK) @ W(NxK)^T  [+ bias[n]] [sigmoid n>=sig_from]
// [+= old Y if accum]. Block = 8 waves; wave computes 16 rows x all N columns.
// ---------------------------------------------------------------------------
template <int N>
__global__ void wmma_gemm_kernel(const float* __restrict__ X, const float* __restrict__ W,
                                 const float* __restrict__ bias, float* __restrict__ Y,
                                 int K, int sig_from, int accum) {
  constexpr int NT = N / 16;
  const int wave = threadIdx.x >> 5, lane = threadIdx.x & 31;
  const int lm = lane & 15, lh = lane >> 4;
  const int m0 = blockIdx.x * 128 + wave * 16;

  v8f acc[NT] = {};
  const float* xrow = X + (size_t)(m0 + lm) * K;

#pragma unroll 2
  for (int k0 = 0; k0 < K; k0 += 4) {
    const int kk = k0 + 2 * lh;
    const v2f a = *(const v2f*)(xrow + kk);
#pragma unroll
    for (int t = 0; t < NT; ++t) {
      const v2f b = *(const v2f*)(W + (size_t)(t * 16 + lm) * K + kk);
      acc[t] = __builtin_amdgcn_wmma_f32_16x16x4_f32(false, a, false, b,
                                                     (short)0, acc[t], false, false);
    }
  }
#pragma unroll
  for (int t = 0; t < NT; ++t) {
    const int n = t * 16 + lm;
    const float badd = bias ? bias[n] : 0.f;
#pragma unroll
    for (int v = 0; v < 8; ++v) {
      const int r = m0 + v + 8 * lh;
      float val = acc[t][v] + badd;
      if (n >= sig_from) val = 1.0f / (1.0f + __expf(-val));
      float* yp = Y + (size_t)r * N + n;
      if (accum) val += *yp;
      *yp = val;
    }
  }
}

// ---------------------------------------------------------------------------
// Fused attention per (head, s):  logits = inv_sqrt_c*(Q K^T + bias),
// softmax over QUERY axis r (per column t), O = g * (A @ V).
// Block = one s, 8 waves; t in panels of 32. Logit panel (384x32) transposed in
// LDS (stride 401 -> conflict-free column reductions). V panel staged to LDS
// with global_load_async_to_lds_b128, drained with s_wait_asynccnt behind
// phase-1 WMMA work. Q fragments register-resident across all panels
// (launch_bounds(256,1) grants the VGPR budget; avoids the round-0 spills).
// ---------------------------------------------------------------------------
__global__ void __launch_bounds__(256, 1)
attn_kernel(const float* __restrict__ qkvg, const float* __restrict__ bgemm,
            float* __restrict__ o_h) {
  __shared__ float lbuf[32 * 401];   // [t][r]  51,328 B
  __shared__ float vbuf[32 * 32];    // [t][c]   4,096 B
  const int s = blockIdx.x;
  const int wave = threadIdx.x >> 5, lane = threadIdx.x & 31;
  const int lm = lane & 15, lh = lane >> 4;
  const int rbase = wave * 48;                 // each wave owns 48 rows (3 M-tiles)
  const float inv_sqrt_c = 0.1767766952966369f;
  const size_t qbase = (size_t)s * R_ * 128;
  const size_t bbase = (size_t)s * P_;

  // async V-panel staging: this thread's 16B chunk
  const int vt = threadIdx.x >> 3;            // t row 0..31
  const int vc = (threadIdx.x & 7) << 2;      // c col {0,4,...,28}
  const unsigned v_lds = lds_off(&vbuf[vt * 32 + vc]);

  // Q fragments: invariant across t-panels; keep register-resident
  v2f qf[8][3];
#pragma unroll
  for (int ks = 0; ks < 8; ++ks) {
    const int kk = ks * 4 + 2 * lh;
#pragma unroll
    for (int mt = 0; mt < 3; ++mt)
      qf[ks][mt] = *(const v2f*)(qkvg + qbase + (size_t)(rbase + mt * 16 + lm) * 128 + kk);
  }

  v8f oacc[3][2] = {};

  for (int tb = 0; tb < 12; ++tb) {
    const int t0 = tb * 32;
    __syncthreads();   // prior panel's LDS readers are done

    // kick off async copy of V panel (32x32 f32) into LDS
    {
      const float* gsrc = qkvg + qbase + (size_t)(t0 + vt) * 128 + 64 + vc;
      asm volatile("global_load_async_to_lds_b128 %0, %1, off"
                   :: "v"(v_lds), "v"(gsrc) : "memory");
    }

    // Phase 1: logits panel (own 48 rows x 32 cols), Q@K^T over K=32
    v8f lacc[3][2] = {};
#pragma unroll
    for (int ks = 0; ks < 8; ++ks) {
      const int kk = ks * 4 + 2 * lh;
      v2f b[2];
#pragma unroll
      for (int nt = 0; nt < 2; ++nt)
        b[nt] = *(const v2f*)(qkvg + qbase + (size_t)(t0 + nt * 16 + lm) * 128 + 32 + kk);
#pragma unroll
      for (int mt = 0; mt < 3; ++mt)
#pragma unroll
        for (int nt = 0; nt < 2; ++nt)
          lacc[mt][nt] = __builtin_amdgcn_wmma_f32_16x16x4_f32(
              false, qf[ks][mt], false, b[nt], (short)0, lacc[mt][nt], false, false);
    }
    // + pair bias, scale, scatter to LDS transposed
#pragma unroll
    for (int mt = 0; mt < 3; ++mt)
#pragma unroll
      for (int nt = 0; nt < 2; ++nt) {
        const int t = nt * 16 + lm;
#pragma unroll
        for (int v = 0; v < 8; ++v) {
          const int r = rbase + mt * 16 + v + 8 * lh;
          const float bb = bgemm[bbase + (size_t)r * R_ + t0 + t];
          lbuf[t * 401 + r] = inv_sqrt_c * (lacc[mt][nt][v] + bb);
        }
      }
    asm volatile("s_wait_asynccnt 0x0" ::: "memory");  // V panel landed (this wave)
    __syncthreads();                                   // ... and block-wide

    // Phase 2: softmax over r (all 384) per column t; wave owns 4 columns
#pragma unroll
    for (int ct = 0; ct < 4; ++ct) {
      float* col = lbuf + (wave * 4 + ct) * 401;
      float m = -3.4e38f;
#pragma unroll
      for (int j = 0; j < 12; ++j) m = fmaxf(m, col[lane + 32 * j]);
      m = wave_max(m);
      float z = 0.f, e[12];
#pragma unroll
      for (int j = 0; j < 12; ++j) { e[j] = __expf(col[lane + 32 * j] - m); z += e[j]; }
      const float rz = 1.0f / wave_sum(z);
#pragma unroll
      for (int j = 0; j < 12; ++j) col[lane + 32 * j] = e[j] * rz;
    }
    __syncthreads();

    // Phase 3: O += A @ V_panel  (A[m=r,k=t] from lbuf, B[k=t,n=c] from vbuf)
#pragma unroll
    for (int ks = 0; ks < 8; ++ks) {
      const int kk = ks * 4 + 2 * lh;
      v2f a[3], b[2];
#pragma unroll
      for (int mt = 0; mt < 3; ++mt) {
        const int r = rbase + mt * 16 + lm;
        a[mt][0] = lbuf[kk * 401 + r];
        a[mt][1] = lbuf[(kk + 1) * 401 + r];
      }
#pragma unroll
      for (int nt = 0; nt < 2; ++nt) {
        const int n = nt * 16 + lm;
        b[nt][0] = vbuf[kk * 32 + n];
        b[nt][1] = vbuf[(kk + 1) * 32 + n];
      }
#pragma unroll
      for (int mt = 0; mt < 3; ++mt)
#pragma unroll
        for (int nt = 0; nt < 2; ++nt)
          oacc[mt][nt] = __builtin_amdgcn_wmma_f32_16x16x4_f32(
              false, a[mt], false, b[nt], (short)0, oacc[mt][nt], false, false);
    }
  }

  // Epilogue: gate with sigmoid(g) (already applied in qkvg), store per-head O
#pragma unroll
  for (int mt = 0; mt < 3; ++mt)
#pragma unroll
    for (int nt = 0; nt < 2; ++nt) {
      const int c = nt * 16 + lm;
#pragma unroll
      for (int v = 0; v < 8; ++v) {
        const int r = rbase + mt * 16 + v + 8 * lh;
        const float g = qkvg[qbase + (size_t)r * 128 + 96 + c];
        o_h[((size_t)s * R_ + r) * C_ + c] = g * oacc[mt][nt][v];
      }
    }
}

// ---------------------------------------------------------------------------
extern "C" void kernel_launch(void* const* d_in, const int* in_sizes, int n_in,
                              void* d_out, int out_size, void* d_ws, size_t ws_size,
                              hipStream_t stream) {
  (void)in_sizes; (void)n_in; (void)out_size; (void)ws_size;
  const float* msa   = (const float*)d_in[0];
  const float* pair  = (const float*)d_in[1];
  const float* nmw   = (const float*)d_in[2];
  const float* nmb   = (const float*)d_in[3];
  const float* wq    = (const float*)d_in[4];
  const float* wk    = (const float*)d_in[5];
  const float* wv    = (const float*)d_in[6];
  const float* npw   = (const float*)d_in[7];
  const float* npb   = (const float*)d_in[8];
  const float* wb    = (const float*)d_in[9];
  const float* wg    = (const float*)d_in[10];
  const float* out_w = (const float*)d_in[11];
  const float* out_b = (const float*)d_in[12];
  float* out = (float*)d_out;

  float* ws = (float*)d_ws;
  float* msa_cur  = ws; ws += (size_t)SR_ * CM_;   // 100.7 MB
  float* phat     = ws; ws += (size_t)P_ * CZ_;    //  75.5 MB
  float* qkvg     = ws; ws += (size_t)SR_ * 128;   //  50.3 MB
  float* bgemm    = ws; ws += (size_t)P_ * S_;     // 151.0 MB (== reshaped bias, flat)
  float* o_h      = ws; ws += (size_t)SR_ * C_;    //  12.6 MB
  float* wcat     = ws; ws += 128 * 256;
  float* wb_eff   = ws; ws += 256 * 128;
  float* bias_eff = ws; ws += 256;
  float* wo_eff   = ws; ws += 256 * 32;

  pair_stats_kernel<<<P_ / 8, 256, 0, stream>>>(pair, phat);

  for (int h = 0; h < H_; ++h) {
    head_prep_kernel<<<289, 256, 0, stream>>>(h, wq, wk, wv, wg, npw, npb, wb, out_w,
                                              wcat, wb_eff, bias_eff, wo_eff);
    ln_msa_kernel<<<SR_ / 8, 256, 0, stream>>>(h == 0 ? msa : msa_cur, msa_cur,
                                               nmw + h * CM_, nmb + h * CM_);
    // q|k|v|sigmoid(g) projection: (SR x 256) @ (128 x 256)^T
    wmma_gemm_kernel<128><<<SR_ / 128, 256, 0, stream>>>(msa_cur, wcat, nullptr, qkvg,
                                                         CM_, 96, 0);
    // pair bias: (P x 128) @ (256 x 128)^T + bias_eff  (flat == b.reshape(S,R,R))
    wmma_gemm_kernel<256><<<P_ / 128, 256, 0, stream>>>(phat, wb_eff, bias_eff, bgemm,
                                                        CZ_, 256, 0);
    attn_kernel<<<S_, 256, 0, stream>>>(qkvg, bgemm, o_h);
    // out (+)= o_h @ out_w[:, h*32:(h+1)*32]^T  (+ out_b on first head)
    wmma_gemm_kernel<256><<<SR_ / 128, 256, 0, stream>>>(o_h, wo_eff,
                                                         h == 0 ? out_b : nullptr,
                                                         out, C_, 256, h > 0 ? 1 : 0);
  }
}